// CapsConv_80599356276737
// MI455X (gfx1250) — compile-verified
//
#include <hip/hip_runtime.h>
#include <hip/hip_bf16.h>
#include <math.h>

#define B_SZ   512
#define K_CAPS 1152
#define J_OUT  10
#define D_IN   8
#define U_OUT  16
#define KK     (K_CAPS * D_IN)   /* 9216 */
#define EPS    1e-7f

typedef __attribute__((ext_vector_type(2))) float v2f;
typedef __attribute__((ext_vector_type(8))) float v8f;

// ---------------- small utility kernels ----------------

__global__ void k_init_b(float* bIJ) {
  int i = blockIdx.x * blockDim.x + threadIdx.x;
  if (i < K_CAPS * J_OUT) bIJ[i] = 0.f;
}

__global__ void k_zero_s(float* s) {
  int i = blockIdx.x * blockDim.x + threadIdx.x;
  if (i < B_SZ * U_OUT) s[i] = 0.f;
}

// c[k] = softmax(b_IJ[k, :])[j]
__global__ void k_softmax_col(const float* __restrict__ bIJ,
                              float* __restrict__ c, int j) {
  int k = blockIdx.x * blockDim.x + threadIdx.x;
  if (k >= K_CAPS) return;
  const float* row = bIJ + k * J_OUT;
  float m = row[0];
#pragma unroll
  for (int t = 1; t < J_OUT; ++t) m = fmaxf(m, row[t]);
  float sum = 0.f;
#pragma unroll
  for (int t = 0; t < J_OUT; ++t) sum += __expf(row[t] - m);
  c[k] = __expf(row[j] - m) / sum;
}

// Wc[kk, u] = c[kk/8] * W[j, kk, u]
__global__ void k_scale_w(const float* __restrict__ W, const float* __restrict__ c,
                          float* __restrict__ Wc, int j) {
  int i = blockIdx.x * blockDim.x + threadIdx.x;
  if (i >= KK * U_OUT) return;
  Wc[i] = c[i >> 7] * W[(size_t)j * (KK * U_OUT) + i];
}

// LDS-tiled transpose: xT[kk, b] = x[b, kk]   (32x32 tiles, 32x8 threads)
__global__ void __launch_bounds__(256)
k_transpose(const float* __restrict__ x, float* __restrict__ xT) {
  __shared__ float tile[32][33];
  const int kk0 = blockIdx.x * 32;
  const int b0  = blockIdx.y * 32;
  const int tx = threadIdx.x, ty = threadIdx.y;
#pragma unroll
  for (int r = 0; r < 32; r += 8)
    tile[ty + r][tx] = x[(size_t)(b0 + ty + r) * KK + kk0 + tx];
  __syncthreads();
#pragma unroll
  for (int r = 0; r < 32; r += 8)
    xT[(size_t)(kk0 + ty + r) * B_SZ + b0 + tx] = tile[tx][ty + r];
}

// v = squash(s) along the 16-vector
__global__ void k_squash(const float* __restrict__ s, float* __restrict__ v) {
  int b = blockIdx.x * blockDim.x + threadIdx.x;
  if (b >= B_SZ) return;
  float tmp[U_OUT];
  float sq = 0.f;
#pragma unroll
  for (int u = 0; u < U_OUT; ++u) { tmp[u] = s[b * U_OUT + u]; sq += tmp[u] * tmp[u]; }
  float scale = (sq / (1.f + sq)) / (sqrtf(sq) + EPS);
#pragma unroll
  for (int u = 0; u < U_OUT; ++u) v[b * U_OUT + u] = tmp[u] * scale;
}

__global__ void k_write_out(const float* __restrict__ v, float* __restrict__ out, int j) {
  int i = blockIdx.x * blockDim.x + threadIdx.x;
  if (i >= B_SZ * U_OUT) return;
  int b = i >> 4, u = i & 15;
  out[b * (J_OUT * U_OUT) + j * U_OUT + u] = v[i];
}

// ---------------- WMMA GEMM kernels ----------------

// Fast path: s[b,u] += sum_kk xT[kk,b] * Wc[kk,u]
// GEMM [512 x 9216] @ [9216 x 16]; all 4 operand loads 16-lane coalesced.
__global__ void __launch_bounds__(32)
k_s_gemm_t(const float* __restrict__ xT, const float* __restrict__ Wc,
           float* __restrict__ s) {
  const int lane = threadIdx.x;
  const int half = lane >> 4;          // 0: K+0/K+1, 1: K+2/K+3
  const int mn   = lane & 15;          // M row (A) / N col (B)
  const int m0   = blockIdx.x * 16;
  const int kk0  = blockIdx.y * 256;

  v8f acc = {};
  for (int kk = kk0; kk < kk0 + 256; kk += 4) {
    const int kq = kk + 2 * half;
    v2f a, b;
    a.x = xT[(size_t)kq * B_SZ + m0 + mn];
    a.y = xT[(size_t)(kq + 1) * B_SZ + m0 + mn];
    b.x = Wc[kq * U_OUT + mn];
    b.y = Wc[(kq + 1) * U_OUT + mn];
    acc = __builtin_amdgcn_wmma_f32_16x16x4_f32(false, a, false, b,
                                                (short)0, acc, false, false);
  }
#pragma unroll
  for (int r = 0; r < 8; ++r) {
    int row = m0 + r + 8 * half;       // C: VGPR r, M = r + 8*half
    unsafeAtomicAdd(&s[row * U_OUT + mn], acc[r]);
  }
}

// Fallback path (small workspace): A from row-major x, c folded inline.
__global__ void __launch_bounds__(32)
k_s_gemm_d(const float* __restrict__ x, const float* __restrict__ W,
           const float* __restrict__ c, float* __restrict__ s, int j) {
  const int lane = threadIdx.x;
  const int half = lane >> 4;
  const int mn   = lane & 15;
  const int m0   = blockIdx.x * 16;
  const int kk0  = blockIdx.y * 256;
  const float* __restrict__ Wj   = W + (size_t)j * (KK * U_OUT);
  const float* __restrict__ arow = x + (size_t)(m0 + mn) * KK;

  v8f acc = {};
  for (int kk = kk0; kk < kk0 + 256; kk += 4) {
    const int kq = kk + 2 * half;
    v2f a, b;
    a.x = arow[kq];
    a.y = arow[kq + 1];
    b.x = c[kq >> 3]       * Wj[kq * U_OUT + mn];
    b.y = c[(kq + 1) >> 3] * Wj[(kq + 1) * U_OUT + mn];
    acc = __builtin_amdgcn_wmma_f32_16x16x4_f32(false, a, false, b,
                                                (short)0, acc, false, false);
  }
#pragma unroll
  for (int r = 0; r < 8; ++r) {
    int row = m0 + r + 8 * half;
    unsafeAtomicAdd(&s[row * U_OUT + mn], acc[r]);
  }
}

// Per capsule pair (k0,k0+1): C[16,16] = x[:, k0*8 : k0*8+16]^T @ v,
// then agree[k] = <C rows for k, W[j,k]>; add into b_IJ[:,j].
// A loads are already coalesced in the row-major x layout.
__global__ void __launch_bounds__(32)
k_agree_gemm(const float* __restrict__ x, const float* __restrict__ W,
             const float* __restrict__ vv, float* __restrict__ bIJ, int j) {
  const int lane = threadIdx.x;
  const int half = lane >> 4;
  const int mn   = lane & 15;
  const int k0      = blockIdx.x * 2;
  const int colbase = k0 * D_IN;       // 16 contiguous kk columns
  const float* __restrict__ Wj = W + (size_t)j * (KK * U_OUT);

  v8f acc = {};
  for (int b0 = 0; b0 < B_SZ; b0 += 4) {
    const int bq = b0 + 2 * half;      // contraction (batch) index
    v2f a, b;
    a.x = x[(size_t)bq * KK + colbase + mn];        // A[m, b] = x[b, colbase+m]
    a.y = x[(size_t)(bq + 1) * KK + colbase + mn];
    b.x = vv[bq * U_OUT + mn];                      // B[b, u] = v[b, u]
    b.y = vv[(bq + 1) * U_OUT + mn];
    acc = __builtin_amdgcn_wmma_f32_16x16x4_f32(false, a, false, b,
                                                (short)0, acc, false, false);
  }
  // acc[r] = C[r + 8*half, mn]; rows 0-7 -> k0, rows 8-15 -> k0+1
  float partial = 0.f;
#pragma unroll
  for (int r = 0; r < 8; ++r) {
    int row = colbase + r + 8 * half;  // flat kk index
    partial += acc[r] * Wj[row * U_OUT + mn];
  }
  for (int off = 8; off > 0; off >>= 1)   // sum over u within each half
    partial += __shfl_xor(partial, off, 16);
  if (mn == 0) {
    int k = k0 + half;                 // exclusive owner of this logit
    bIJ[k * J_OUT + j] += partial;
  }
}

// ---------------- orchestration ----------------

extern "C" void kernel_launch(void* const* d_in, const int* in_sizes, int n_in,
                              void* d_out, int out_size, void* d_ws, size_t ws_size,
                              hipStream_t stream) {
  const float* x = (const float*)d_in[0];   // [B, K, 8, 1] -> [B, 9216]
  const float* W = (const float*)d_in[1];   // [10, K, 8, 16]
  float* out = (float*)d_out;               // [B, 10, 16, 1]

  float* bIJ = (float*)d_ws;                        // K*J        (11520)
  float* c   = bIJ + K_CAPS * J_OUT;                // K          (1152)
  float* s   = c   + K_CAPS;                        // B*16       (8192)
  float* v   = s   + B_SZ * U_OUT;                  // B*16       (8192)
  float* Wc  = v   + B_SZ * U_OUT;                  // KK*16      (147456)
  float* xT  = Wc  + KK * U_OUT;                    // KK*B       (4718592)

  const size_t need_full =
      (size_t)(K_CAPS * J_OUT + K_CAPS + 2 * B_SZ * U_OUT + KK * U_OUT
               + (size_t)KK * B_SZ) * sizeof(float);
  const bool fast = (ws_size >= need_full);

  k_init_b<<<dim3((K_CAPS * J_OUT + 255) / 256), 256, 0, stream>>>(bIJ);
  if (fast) {
    k_transpose<<<dim3(KK / 32, B_SZ / 32), dim3(32, 8), 0, stream>>>(x, xT);
  }

  for (int j = 0; j < J_OUT; ++j) {
    for (int it = 0; it < 3; ++it) {
      k_softmax_col<<<dim3((K_CAPS + 255) / 256), 256, 0, stream>>>(bIJ, c, j);
      k_zero_s<<<dim3((B_SZ * U_OUT + 255) / 256), 256, 0, stream>>>(s);
      if (fast) {
        k_scale_w<<<dim3((KK * U_OUT + 255) / 256), 256, 0, stream>>>(W, c, Wc, j);
        k_s_gemm_t<<<dim3(B_SZ / 16, KK / 256), 32, 0, stream>>>(xT, Wc, s);
      } else {
        k_s_gemm_d<<<dim3(B_SZ / 16, KK / 256), 32, 0, stream>>>(x, W, c, s, j);
      }
      k_squash<<<dim3((B_SZ + 255) / 256), 256, 0, stream>>>(s, v);
      k_agree_gemm<<<dim3(K_CAPS / 2), 32, 0, stream>>>(x, W, v, bIJ, j);
    }
    k_write_out<<<dim3((B_SZ * U_OUT + 255) / 256), 256, 0, stream>>>(v, out, j);
  }
}